// PrototypeClassifier_63702954934921
// MI455X (gfx1250) — compile-verified
//
#include <hip/hip_runtime.h>

// Problem constants (match reference)
#define B_N 8192
#define P_N 2048
#define L_N 1024
#define O_N 1000

typedef __attribute__((ext_vector_type(2))) float v2f;
typedef __attribute__((ext_vector_type(8))) float v8f;

// ---------------------------------------------------------------------------
// Kernel 1: partial column sums + partial Frobenius-norm of prototypes.
// grid = (L/256, 64 chunks of 32 rows), block = 256.
// part_col[chunk*L + l] = sum_{p in chunk} proto[p,l]
// part_sq[block]        = sum_{p in chunk, l in block-cols} proto^2
// ---------------------------------------------------------------------------
__global__ void proto_partial_kernel(const float* __restrict__ proto,
                                     float* __restrict__ part_col,
                                     float* __restrict__ part_sq) {
  __shared__ float sdata[256];
  const int tid = threadIdx.x;
  const int l = blockIdx.x * 256 + tid;
  const int p0 = blockIdx.y * 32;
  float acc = 0.f, accsq = 0.f;
#pragma unroll 8
  for (int i = 0; i < 32; ++i) {
    float v = proto[(size_t)(p0 + i) * L_N + l];
    acc += v;
    accsq += v * v;
  }
  part_col[blockIdx.y * L_N + l] = acc;
  sdata[tid] = accsq;
  __syncthreads();
  for (int s = 128; s > 0; s >>= 1) {
    if (tid < s) sdata[tid] += sdata[tid + s];
    __syncthreads();
  }
  if (tid == 0) part_sq[blockIdx.y * gridDim.x + blockIdx.x] = sdata[0];
}

// ---------------------------------------------------------------------------
// Kernel 2: finalize colsum[L] and the scalar ||prototypes||_F^2.
// grid = 4, block = 256. Fixed-order sums -> deterministic.
// ---------------------------------------------------------------------------
__global__ void finalize_kernel(const float* __restrict__ part_col,
                                const float* __restrict__ part_sq,
                                float* __restrict__ colsum,
                                float* __restrict__ sq) {
  const int l = blockIdx.x * 256 + threadIdx.x;
  float acc = 0.f;
  for (int c = 0; c < 64; ++c) acc += part_col[c * L_N + l];
  colsum[l] = acc;
  if (blockIdx.x == 0 && threadIdx.x == 0) {
    float s = 0.f;
    for (int i = 0; i < 256; ++i) s += part_sq[i];
    *sq = s;
  }
}

// ---------------------------------------------------------------------------
// Kernel 3: d[b] = sqrt(P*||x_b||^2 - 2*x_b.colsum + proto_sq)
// One wave32 per row; 8 rows per 256-thread block. colsum staged in LDS.
// grid = B/8 = 1024 blocks.
// ---------------------------------------------------------------------------
__global__ void dist_kernel(const float* __restrict__ x,
                            const float* __restrict__ colsum,
                            const float* __restrict__ sqp,
                            float* __restrict__ dvec) {
  __shared__ float scol[L_N];
  const int tid = threadIdx.x;
#pragma unroll
  for (int i = 0; i < L_N / 256; ++i) scol[tid + 256 * i] = colsum[tid + 256 * i];
  __syncthreads();
  const int lane = tid & 31;
  const int wave = tid >> 5;
  const int row = blockIdx.x * 8 + wave;
  const float* xr = x + (size_t)row * L_N;
  float xx = 0.f, xc = 0.f;
#pragma unroll 8
  for (int i = 0; i < 32; ++i) {
    const int idx = lane + 32 * i;  // coalesced across the wave
    float v = xr[idx];
    xx += v * v;
    xc += v * scol[idx];
  }
#pragma unroll
  for (int m = 16; m >= 1; m >>= 1) {
    xx += __shfl_xor(xx, m, 32);
    xc += __shfl_xor(xc, m, 32);
  }
  if (lane == 0) {
    float s = (float)P_N * xx - 2.f * xc + *sqp;
    dvec[row] = sqrtf(s);
  }
}

// ---------------------------------------------------------------------------
// Kernel 4: wsum[o] = sum_p weight[o,p]. One wave per row, 8 rows/block.
// grid = O/8 = 125 blocks (1000 = 125*8 exactly).
// ---------------------------------------------------------------------------
__global__ void wsum_kernel(const float* __restrict__ weight,
                            float* __restrict__ wsum) {
  const int tid = threadIdx.x;
  const int lane = tid & 31;
  const int wave = tid >> 5;
  const int o = blockIdx.x * 8 + wave;
  const float* wr = weight + (size_t)o * P_N;
  float acc = 0.f;
#pragma unroll 8
  for (int i = 0; i < P_N / 32; ++i) acc += wr[lane + 32 * i];
#pragma unroll
  for (int m = 16; m >= 1; m >>= 1) acc += __shfl_xor(acc, m, 32);
  if (lane == 0) wsum[o] = acc;
}

// ---------------------------------------------------------------------------
// Kernel 5: out[b,o] = d[b]*wsum[o] + bias[o] as 16x16 tiles via
// V_WMMA_F32_16X16X4_F32 (rank-1: only K=0 of A nonzero, bias preloaded in C).
// One tile per wave, 8 waves per block. Tiles = (B/16)*ceil(O/16) = 512*63.
// ---------------------------------------------------------------------------
__global__ void outer_wmma_kernel(const float* __restrict__ dvec,
                                  const float* __restrict__ wsum,
                                  const float* __restrict__ bias,
                                  float* __restrict__ out) {
  const int tid = threadIdx.x;
  const int lane = tid & 31;
  const int wave = tid >> 5;
  const int tile = blockIdx.x * 8 + wave;
  const int OT = (O_N + 15) / 16;  // 63 tiles along O
  const int bt = tile / OT;
  const int ot = tile - bt * OT;
  const int b0 = bt * 16;
  const int o0 = ot * 16;

  const int n = lane & 15;
  const int ocl = min(o0 + n, O_N - 1);  // clamp for the ragged last tile
  const float biasv = bias[ocl];

  // A (16x4 f32): lanes 0-15, VGPR0 = K=0 column -> d values; all else zero.
  v2f a;
  a.x = (lane < 16) ? dvec[b0 + lane] : 0.f;
  a.y = 0.f;
  // B (4x16 f32): lanes 0-15, VGPR0 = K=0 row -> wsum values; all else zero.
  v2f b;
  b.x = (lane < 16) ? wsum[ocl] : 0.f;
  b.y = 0.f;
  // C (16x16 f32): every row gets bias[o0 + n]; C/D layout N = lane%16.
  v8f c;
#pragma unroll
  for (int r = 0; r < 8; ++r) c[r] = biasv;

  // D = A x B + C  -> d[b]*wsum[o] + bias[o]
  v8f d = __builtin_amdgcn_wmma_f32_16x16x4_f32(
      /*neg_a=*/false, a, /*neg_b=*/false, b,
      /*c_mod=*/(short)0, c, /*reuse_a=*/false, /*reuse_b=*/false);

  // D layout: VGPR r -> M=r (lanes 0-15) or M=r+8 (lanes 16-31), N=lane%16.
  const int mbase = (lane < 16) ? 0 : 8;
  if (o0 + n < O_N) {
#pragma unroll
    for (int r = 0; r < 8; ++r) {
      out[(size_t)(b0 + mbase + r) * O_N + (o0 + n)] = d[r];
    }
  }
}

// ---------------------------------------------------------------------------
// Workspace layout (floats):
//   [0      .. 65535] part_col  (64 x 1024)
//   [65536  .. 65791] part_sq   (256)
//   [65792  .. 66815] colsum    (1024)
//   [66816]           proto_sq  (1)
//   [66824  .. 75015] dvec      (8192)
//   [75016  .. 76015] wsum      (1000)
// Total ~304 KB.
// ---------------------------------------------------------------------------
extern "C" void kernel_launch(void* const* d_in, const int* in_sizes, int n_in,
                              void* d_out, int out_size, void* d_ws, size_t ws_size,
                              hipStream_t stream) {
  const float* inp    = (const float*)d_in[0];  // [B, L]
  const float* proto  = (const float*)d_in[1];  // [P, L]
  const float* weight = (const float*)d_in[2];  // [O, P]
  const float* bias   = (const float*)d_in[3];  // [O]
  float* out = (float*)d_out;                   // [B, O]

  float* ws       = (float*)d_ws;
  float* part_col = ws;
  float* part_sq  = ws + 65536;
  float* colsum   = ws + 65792;
  float* sq       = ws + 66816;
  float* dvec     = ws + 66824;
  float* wsum     = ws + 75016;

  hipLaunchKernelGGL(proto_partial_kernel, dim3(L_N / 256, 64), dim3(256), 0, stream,
                     proto, part_col, part_sq);
  hipLaunchKernelGGL(finalize_kernel, dim3(L_N / 256), dim3(256), 0, stream,
                     part_col, part_sq, colsum, sq);
  hipLaunchKernelGGL(dist_kernel, dim3(B_N / 8), dim3(256), 0, stream,
                     inp, colsum, sq, dvec);
  hipLaunchKernelGGL(wsum_kernel, dim3(O_N / 8), dim3(256), 0, stream,
                     weight, wsum);
  const int tiles = (B_N / 16) * ((O_N + 15) / 16);  // 512 * 63 = 32256
  hipLaunchKernelGGL(outer_wmma_kernel, dim3(tiles / 8), dim3(256), 0, stream,
                     dvec, wsum, bias, out);
}